// SSIM_80702435491974
// MI455X (gfx1250) — compile-verified
//
#include <hip/hip_runtime.h>

// CDNA5 (gfx1250) SSIM: separable 11-tap Gaussian convs done as banded-Toeplitz
// matmuls on V_WMMA_F32_16X16X4_F32 (fp32-preserving), wave32, one wave = one
// 16x16 output tile.

typedef __attribute__((ext_vector_type(2))) float v2f;
typedef __attribute__((ext_vector_type(8))) float v8f;

#define H_IMG 1024
#define W_IMG 1024
#define OUT_H 1014   // 1024 - 11 + 1
#define OUT_W 1014
#define NIMG  32

static __device__ __forceinline__ v8f wmma4(v2f a, v2f b, v8f c) {
  // 8 args: (neg_a, A, neg_b, B, c_mod, C, reuse_a, reuse_b)
  return __builtin_amdgcn_wmma_f32_16x16x4_f32(
      false, a, false, b, (short)0, c, false, false);
}

__global__ void ssim_init_ws(float* ws) {
  int i = threadIdx.x;
  if (i < 3 * NIMG) ws[i] = 0.0f;
}

__global__ __launch_bounds__(32)
void ssim_tile_kernel(const float* __restrict__ img1,
                      const float* __restrict__ img2,
                      const float* __restrict__ win,   // 11x11 = g * g^T
                      float* __restrict__ ws) {
  __shared__ float g_s[16];
  __shared__ float Hbuf[5][32][17];   // 5 signals, 32 intermediate rows, pad 17

  const int lane = threadIdx.x;            // 0..31 (wave32)
  const int b    = blockIdx.z;             // image
  const int r0   = blockIdx.y * 16;        // output tile origin
  const int c0   = blockIdx.x * 16;

  // Recover the separable 1-D Gaussian: row-sums of window (rows sum to g_i).
  if (lane < 11) {
    float s = 0.f;
    #pragma unroll
    for (int j = 0; j < 11; ++j) s += win[lane * 11 + j];
    g_s[lane] = s;
  }
  __syncthreads();

  const int n  = lane & 15;          // A-layout row m / B-layout col n
  const int k0 = (lane >> 4) * 2;    // K sub-index held by this half-wave

  // Banded Toeplitz weight fragments W[j, c] = g[j - c], chunked K=4.
  // Identical fragment serves as B in the H-pass and A in the V-pass.
  v2f wfrag[7];
  #pragma unroll
  for (int t = 0; t < 7; ++t) {
    int i0 = 4 * t + k0 - n;
    int i1 = i0 + 1;
    wfrag[t].x = (i0 >= 0 && i0 <= 10) ? g_s[i0] : 0.f;
    wfrag[t].y = (i1 >= 0 && i1 <= 10) ? g_s[i1] : 0.f;
  }

  const float* p1 = img1 + (size_t)b * (H_IMG * (size_t)W_IMG);
  const float* p2 = img2 + (size_t)b * (H_IMG * (size_t)W_IMG);

  // ---------------- horizontal pass: 32 intermediate rows x 16 cols ----------
  #pragma unroll
  for (int rb = 0; rb < 32; rb += 16) {
    v8f h0 = {}; v8f h1 = {}; v8f h2 = {}; v8f h3 = {}; v8f h4 = {};
    int row = r0 + rb + n; if (row > H_IMG - 1) row = H_IMG - 1;
    const float2* r1 = (const float2*)(p1 + (size_t)row * W_IMG);
    const float2* r2 = (const float2*)(p2 + (size_t)row * W_IMG);
    #pragma unroll
    for (int t = 0; t < 7; ++t) {
      int col = c0 + 4 * t + k0;                // always even -> 8B aligned
      if (col > W_IMG - 2) col = W_IMG - 2;     // clamp feeds zero-weight taps
      float2 x1 = r1[col >> 1];
      float2 x2 = r2[col >> 1];
      v2f a1;  a1.x  = x1.x;       a1.y  = x1.y;
      v2f a2;  a2.x  = x2.x;       a2.y  = x2.y;
      v2f a11 = a1 * a1;
      v2f a22 = a2 * a2;
      v2f a12 = a1 * a2;
      h0 = wmma4(a1,  wfrag[t], h0);   // mu1 partial
      h1 = wmma4(a2,  wfrag[t], h1);   // mu2 partial
      h2 = wmma4(a11, wfrag[t], h2);   // E[x^2] partial
      h3 = wmma4(a22, wfrag[t], h3);   // E[y^2] partial
      h4 = wmma4(a12, wfrag[t], h4);   // E[xy] partial
    }
    // C-layout -> LDS (row-major) so the V-pass can fetch B-layout fragments.
    #pragma unroll
    for (int v = 0; v < 8; ++v) {
      int y = rb + v + 8 * (lane >> 4);
      Hbuf[0][y][n] = h0[v];
      Hbuf[1][y][n] = h1[v];
      Hbuf[2][y][n] = h2[v];
      Hbuf[3][y][n] = h3[v];
      Hbuf[4][y][n] = h4[v];
    }
  }
  __syncthreads();

  // ---------------- vertical pass: Toeplitz(A) x intermediate(B) ------------
  v8f d0 = {}; v8f d1 = {}; v8f d2 = {}; v8f d3 = {}; v8f d4 = {};
  #pragma unroll
  for (int t = 0; t < 7; ++t) {
    int yy = 4 * t + k0;                        // <= 27 < 32
    v2f b0; b0.x = Hbuf[0][yy][n]; b0.y = Hbuf[0][yy + 1][n];
    v2f b1; b1.x = Hbuf[1][yy][n]; b1.y = Hbuf[1][yy + 1][n];
    v2f b2; b2.x = Hbuf[2][yy][n]; b2.y = Hbuf[2][yy + 1][n];
    v2f b3; b3.x = Hbuf[3][yy][n]; b3.y = Hbuf[3][yy + 1][n];
    v2f b4; b4.x = Hbuf[4][yy][n]; b4.y = Hbuf[4][yy + 1][n];
    d0 = wmma4(wfrag[t], b0, d0);
    d1 = wmma4(wfrag[t], b1, d1);
    d2 = wmma4(wfrag[t], b2, d2);
    d3 = wmma4(wfrag[t], b3, d3);
    d4 = wmma4(wfrag[t], b4, d4);
  }

  // ---------------- pointwise SSIM + tile reduction --------------------------
  const float C1 = 0.01f * 0.01f;               // (0.01*L)^2, L=1
  const float C2 = 0.03f * 0.03f;
  float sum_ssim = 0.f, sum_cs = 0.f, sum_sig = 0.f;
  #pragma unroll
  for (int v = 0; v < 8; ++v) {
    int r = r0 + v + 8 * (lane >> 4);           // C/D layout -> pixel coords
    int c = c0 + n;
    if (r < OUT_H && c < OUT_W) {
      float mu1  = d0[v], mu2 = d1[v];
      float mu1s = mu1 * mu1, mu2s = mu2 * mu2, mu12 = mu1 * mu2;
      float s1 = d2[v] - mu1s; s1 = s1 > 0.f ? s1 : 0.f;
      float s2 = d3[v] - mu2s; s2 = s2 > 0.f ? s2 : 0.f;
      float s12 = d4[v] - mu12;
      float m1 = 2.f * mu12 + C1;
      float m2 = mu1s + mu2s + C1;
      float v1 = 2.f * s12 + C2;
      float v2 = s1 + s2 + C2;
      float cs = v1 / v2;
      float ss = (m1 * v1) / (m2 * v2);
      float sg = s1 > 1e-4f ? s1 : 1e-4f;
      sum_ssim += ss; sum_cs += cs; sum_sig += sg;
    }
  }
  // wave32 butterfly reduction
  #pragma unroll
  for (int off = 16; off > 0; off >>= 1) {
    sum_ssim += __shfl_xor(sum_ssim, off, 32);
    sum_cs   += __shfl_xor(sum_cs,   off, 32);
    sum_sig  += __shfl_xor(sum_sig,  off, 32);
  }
  if (lane == 0) {
    atomicAdd(&ws[b],            sum_ssim);
    atomicAdd(&ws[NIMG + b],     sum_cs);
    atomicAdd(&ws[2 * NIMG + b], sum_sig);
  }
}

__global__ void ssim_finalize(const float* __restrict__ ws,
                              float* __restrict__ out) {
  int i = threadIdx.x;
  if (i < 3 * NIMG)
    out[i] = ws[i] * (1.0f / ((float)OUT_H * (float)OUT_W));
}

extern "C" void kernel_launch(void* const* d_in, const int* in_sizes, int n_in,
                              void* d_out, int out_size, void* d_ws, size_t ws_size,
                              hipStream_t stream) {
  (void)in_sizes; (void)n_in; (void)out_size; (void)ws_size;
  const float* img1 = (const float*)d_in[0];   // [32,1,1024,1024] f32
  const float* img2 = (const float*)d_in[1];   // [32,1,1024,1024] f32
  const float* win  = (const float*)d_in[2];   // [1,1,11,11] f32
  float* out = (float*)d_out;                  // 96 f32: ssim | cs | sigma
  float* ws  = (float*)d_ws;                   // 96 f32 accumulators

  ssim_init_ws<<<1, 128, 0, stream>>>(ws);
  dim3 grid((OUT_W + 15) / 16, (OUT_H + 15) / 16, NIMG);  // 64 x 64 x 32
  ssim_tile_kernel<<<grid, 32, 0, stream>>>(img1, img2, win, ws);
  ssim_finalize<<<1, 128, 0, stream>>>(ws, out);
}